// Periodic_13176959664669
// MI455X (gfx1250) — compile-verified
//
#include <hip/hip_runtime.h>
#include <math.h>

typedef float v2f __attribute__((ext_vector_type(2)));
typedef float v8f __attribute__((ext_vector_type(8)));

#define MAX_CELL_MULTIPLES 100000

// ---------------------------------------------------------------------------
// Prologue: replicate the host-side _grid_bounds() on device (1 thread).
// Writes {a_max, b_max, c_max} into ws[0..2].
// ---------------------------------------------------------------------------
__global__ void Periodic_bounds_kernel(const float* __restrict__ cell,
                                       const float* __restrict__ cutoffp,
                                       const float* __restrict__ r1,
                                       const float* __restrict__ r2,
                                       int* __restrict__ bounds) {
    if (threadIdx.x != 0 || blockIdx.x != 0) return;

    float a[3] = {cell[0], cell[1], cell[2]};
    float b[3] = {cell[3], cell[4], cell[5]};
    float c[3] = {cell[6], cell[7], cell[8]};
    float dr[3] = {r2[0] - r1[0], r2[1] - r1[1], r2[2] - r1[2]};
    float cutoff = cutoffp[0];

    float bxc[3] = {b[1]*c[2] - b[2]*c[1],
                    b[2]*c[0] - b[0]*c[2],
                    b[0]*c[1] - b[1]*c[0]};
    float vol = fabsf(a[0]*bxc[0] + a[1]*bxc[1] + a[2]*bxc[2]);

    const float* U[3] = {b, a, a};
    const float* V[3] = {c, c, b};
    for (int q = 0; q < 3; ++q) {
        const float* u = U[q];
        const float* v = V[q];
        float cr[3] = {u[1]*v[2] - u[2]*v[1],
                       u[2]*v[0] - u[0]*v[2],
                       u[0]*v[1] - u[1]*v[0]};
        float ln = sqrtf(cr[0]*cr[0] + cr[1]*cr[1] + cr[2]*cr[2]);
        float d  = fabsf(dr[0]*cr[0] + dr[1]*cr[1] + dr[2]*cr[2]) / ln;
        int nm = (int)floorf((cutoff + d) / vol * ln);
        bounds[q] = nm < MAX_CELL_MULTIPLES ? nm : MAX_CELL_MULTIPLES;
    }
}

// ---------------------------------------------------------------------------
// Main kernel: each wave32 processes a 16-row tile of the [N,3] problem with
// one V_WMMA_F32_16X16X4_F32:  D(16x16) = A(16x4: i,j,k,0) * B(4x16: cell
// padded) + C(dr broadcast).  D is staged through LDS; lanes 0-15 gather
// their row's 3 components, apply the cutoff mask, and store.
// ---------------------------------------------------------------------------
__global__ __launch_bounds__(256)
void Periodic_13176959664669_kernel(const float* __restrict__ cell,
                                    const float* __restrict__ cutoffp,
                                    const float* __restrict__ r1,
                                    const float* __restrict__ r2,
                                    const int* __restrict__ bounds,
                                    float* __restrict__ out,
                                    int N) {
    __shared__ float tile[8][16][16];   // per-wave 16x16 D staging (8 KB)

    const int lane = threadIdx.x & 31;
    const int wave = threadIdx.x >> 5;
    const int halfsel = lane >> 4;      // 0: lanes 0-15, 1: lanes 16-31
    const int lp = lane & 15;

    const int a_max = bounds[0];
    const int b_max = bounds[1];
    const int c_max = bounds[2];
    const int DB = 2 * b_max + 1;
    const int DC = 2 * c_max + 1;
    const int DBC = DB * DC;

    const float cut  = cutoffp[0];
    const float cut2 = cut * cut;
    const float drn0 = r2[0] - r1[0];
    const float drn1 = r2[1] - r1[1];
    const float drn2 = r2[2] - r1[2];

    // ---- B matrix (4x16, K x N): VGPR v <-> K=v (lanes 0-15), K=v+2 (16-31).
    // B[k][n] = cell[k*3 + n] for n < 3, else 0.  K=3 row is zero padding.
    v2f B;
    if (lp < 3) {
        B.x = halfsel ? cell[2 * 3 + lp] : cell[0 * 3 + lp];
        B.y = halfsel ? 0.0f             : cell[1 * 3 + lp];
    } else {
        B.x = 0.0f; B.y = 0.0f;
    }

    // ---- C matrix: C[m][n] = dr[n] so D = ijk@cell + dr directly.
    float drv = 0.0f;
    if (lp == 0) drv = drn0;
    else if (lp == 1) drv = drn1;
    else if (lp == 2) drv = drn2;
    v8f C;
#pragma unroll
    for (int r = 0; r < 8; ++r) C[r] = drv;

    float* vec_out  = out;
    int*   idx_out  = (int*)(out + (size_t)3 * N);
    float* mask_out = out + (size_t)6 * N;

    const int tiles = (N + 15) >> 4;
    const int wave_stride = gridDim.x * 8;

    for (int t = blockIdx.x * 8 + wave; t < tiles; t += wave_stride) {
        const int row = t * 16 + lp;            // this lane's matrix row M
        const int rr  = row < N ? row : N - 1;  // clamp compute, guard stores

        // Decode lattice triple (i outer, k inner — matches meshgrid 'ij').
        const int i   = rr / DBC - a_max;
        const int rem = rr % DBC;
        const int j   = rem / DC - b_max;
        const int k   = rem % DC - c_max;

        // ---- A matrix (16x4): lanes 0-15 hold K=0,1; lanes 16-31 hold K=2,3.
        v2f A;
        A.x = halfsel ? (float)k : (float)i;
        A.y = halfsel ? 0.0f     : (float)j;

        // D = A*B + C  (only columns 0-2 of D are meaningful)
        v8f D = __builtin_amdgcn_wmma_f32_16x16x4_f32(
            false, A, false, B, (short)0, C, false, false);

        // Stage D into LDS: VGPR r holds row r (lanes 0-15) / row r+8 (16-31).
#pragma unroll
        for (int r = 0; r < 8; ++r)
            tile[wave][r + halfsel * 8][lp] = D[r];

        // Intra-wave LDS RAW ordering (CDNA5 split DS counter).
        asm volatile("s_wait_dscnt 0" ::: "memory");

        // Lanes 0-15: gather row lp's components, mask, store.
        if (!halfsel && row < N) {
            const float v0 = tile[wave][lp][0];
            const float v1 = tile[wave][lp][1];
            const float v2 = tile[wave][lp][2];
            const bool m = (v0 * v0 + v1 * v1 + v2 * v2) < cut2;

            const size_t o3 = (size_t)row * 3;
            vec_out[o3 + 0] = m ? v0 : 0.0f;
            vec_out[o3 + 1] = m ? v1 : 0.0f;
            vec_out[o3 + 2] = m ? v2 : 0.0f;
            idx_out[o3 + 0] = m ? i : 0;
            idx_out[o3 + 1] = m ? j : 0;
            idx_out[o3 + 2] = m ? k : 0;
            mask_out[row]   = m ? 1.0f : 0.0f;
        }
        // EXEC restored to all-ones here; next WMMA iteration is safe.
    }
}

extern "C" void kernel_launch(void* const* d_in, const int* in_sizes, int n_in,
                              void* d_out, int out_size, void* d_ws, size_t ws_size,
                              hipStream_t stream) {
    const float* cell   = (const float*)d_in[0];  // [3,3] row-major
    const float* cutoff = (const float*)d_in[1];  // scalar
    const float* r1     = (const float*)d_in[2];  // [3]
    const float* r2     = (const float*)d_in[3];  // [3]

    int* bounds = (int*)d_ws;                     // {a_max, b_max, c_max}

    // out = vectors[N*3] ++ cell_indices[N*3] ++ mask[N]  =>  N = out_size / 7
    const int N = out_size / 7;

    Periodic_bounds_kernel<<<1, 32, 0, stream>>>(cell, cutoff, r1, r2, bounds);

    const int tiles  = (N + 15) >> 4;             // 16 rows per wave
    int blocks = (tiles + 7) / 8;                 // 8 waves per 256-thread block
    if (blocks < 1) blocks = 1;

    Periodic_13176959664669_kernel<<<blocks, 256, 0, stream>>>(
        cell, cutoff, r1, r2, bounds, (float*)d_out, N);
}